// ASAP_58033598104020
// MI455X (gfx1250) — compile-verified
//
#include <hip/hip_runtime.h>
#include <hip/hip_bf16.h>
#include <math.h>

// ---------------------------------------------------------------------------
// ASAP GNN edge-MLP + scatter-max for MI455X (gfx1250).
//   E = 1.6M edges, N = 100K nodes, HIDDEN = 64.
//   Layers 2/3 are E x 64 @ 64 x 64 f32 GEMMs -> V_WMMA_F32_16X16X4_F32.
//   BatchNorm of layer k is folded into the weights of layer k+1 on device.
// ---------------------------------------------------------------------------

typedef __attribute__((ext_vector_type(2))) float v2f;
typedef __attribute__((ext_vector_type(8))) float v8f;

#define BN_EPS 1e-5f

// ---------------- f32 atomic max on global memory (CDNA5 native) -----------
__device__ __forceinline__ void atomic_max_f32(float* addr, float val) {
#if defined(__gfx1250__)
  asm volatile("global_atomic_max_num_f32 %0, %1, off scope:SCOPE_DEV"
               :: "v"(addr), "v"(val)
               : "memory");
#else
  (void)atomicMax(addr, val);
#endif
}

// ---------------- init: zero accumulators, fill output with -inf -----------
__global__ void ASAP_init_kernel(float* ws, float* out, long out_n) {
  long tid = (long)blockIdx.x * blockDim.x + threadIdx.x;
  if (tid < 1024) ws[tid] = 0.0f;                       // normsq + all stats
  if (tid < out_n) out[tid] = __int_as_float(0xff800000); // -inf
}

// ---------------- global scalar norm of all edge directions ----------------
__global__ __launch_bounds__(256) void ASAP_norm_kernel(
    const float* __restrict__ pos, const int* __restrict__ ei,
    float* __restrict__ normsq, int E) {
  __shared__ float red;
  if (threadIdx.x == 0) red = 0.0f;
  __syncthreads();
  long e = (long)blockIdx.x * 256 + threadIdx.x;
  float acc = 0.0f;
  if (e < E) {
    int s = ei[e], d = ei[(long)E + e];
    float dx = pos[3 * s + 0] - pos[3 * d + 0];
    float dy = pos[3 * s + 1] - pos[3 * d + 1];
    float dz = pos[3 * s + 2] - pos[3 * d + 2];
    acc = dx * dx + dy * dy + dz * dz;
  }
  #pragma unroll
  for (int m = 16; m > 0; m >>= 1) acc += __shfl_xor(acc, m, 32);
  if ((threadIdx.x & 31) == 0) atomicAdd(&red, acc);
  __syncthreads();
  if (threadIdx.x == 0) atomicAdd(normsq, red);
}

// ---------------- layer 1: gather + [E,6]@[6,64] + ReLU + stats ------------
__global__ __launch_bounds__(256) void ASAP_layer1_kernel(
    const float* __restrict__ x, const float* __restrict__ pos,
    const int* __restrict__ ei, const float* __restrict__ W1,
    const float* __restrict__ b1, const float* __restrict__ normsq,
    float* __restrict__ H, float* __restrict__ gSum, float* __restrict__ gSq,
    int E) {
  __shared__ float sW[6 * 64];
  __shared__ float sb[64];
  __shared__ float sSum[64];
  __shared__ float sSq[64];
  int t = threadIdx.x;
  for (int i = t; i < 384; i += 256) sW[i] = W1[i];
  if (t < 64) { sb[t] = b1[t]; sSum[t] = 0.0f; sSq[t] = 0.0f; }
  __syncthreads();

  long e = (long)blockIdx.x * 256 + t;
  bool valid = e < E;
  float h[64];
  if (valid) {
    int s = ei[e], d = ei[(long)E + e];
    float rinv = rsqrtf(normsq[0]);   // 1 / ||all directions||  (global scalar)
    float f0 = (pos[3 * s + 0] - pos[3 * d + 0]) * rinv;
    float f1 = (pos[3 * s + 1] - pos[3 * d + 1]) * rinv;
    float f2 = (pos[3 * s + 2] - pos[3 * d + 2]) * rinv;
    float f3 = x[3 * d + 0];
    float f4 = x[3 * d + 1];
    float f5 = x[3 * d + 2];
    #pragma unroll
    for (int c = 0; c < 64; ++c) {
      float v = sb[c];
      v = fmaf(f0, sW[0 * 64 + c], v);
      v = fmaf(f1, sW[1 * 64 + c], v);
      v = fmaf(f2, sW[2 * 64 + c], v);
      v = fmaf(f3, sW[3 * 64 + c], v);
      v = fmaf(f4, sW[4 * 64 + c], v);
      v = fmaf(f5, sW[5 * 64 + c], v);
      h[c] = fmaxf(v, 0.0f);          // ReLU (pre-BN activation, stored)
    }
    float* dst = H + e * 64;
    #pragma unroll
    for (int q = 0; q < 16; ++q)
      *(float4*)(dst + 4 * q) = make_float4(h[4 * q], h[4 * q + 1],
                                            h[4 * q + 2], h[4 * q + 3]);
    // conflict-free rotated LDS accumulation of per-channel sum / sumsq
    int lane = t & 31;
    #pragma unroll
    for (int j = 0; j < 64; ++j) {
      int c = ((lane << 1) + j) & 63;
      float v = h[c];
      atomicAdd(&sSum[c], v);
      atomicAdd(&sSq[c], v * v);
    }
  }
  __syncthreads();
  if (t < 64) { atomicAdd(&gSum[t], sSum[t]); atomicAdd(&gSq[t], sSq[t]); }
}

// ---------------- fold BN(prev layer) into next layer's weights ------------
// a[k] = g[k]*rsqrt(var[k]+eps);  c[k] = be[k] - mu[k]*a[k]
// W'[k][n] = a[k]*W[k][n];  b'[n] = b[n] + sum_k c[k]*W[k][n]
__global__ void ASAP_fold_kernel(
    const float* __restrict__ sum, const float* __restrict__ sq,
    const float* __restrict__ g, const float* __restrict__ be,
    const float* __restrict__ Wnext, const float* __restrict__ bnext,
    float* __restrict__ Wf, float* __restrict__ bf,
    float* __restrict__ aOut, float* __restrict__ cOut, float invE) {
  __shared__ float a[64];
  __shared__ float c[64];
  int t = threadIdx.x;  // 64 threads
  float mu = sum[t] * invE;
  float var = sq[t] * invE - mu * mu;
  float av = g[t] * rsqrtf(var + BN_EPS);
  float cv = be[t] - mu * av;
  a[t] = av;
  c[t] = cv;
  if (aOut) { aOut[t] = av; cOut[t] = cv; }
  __syncthreads();
  if (Wf) {
    float acc = bnext[t];
    for (int k = 0; k < 64; ++k) {
      float w = Wnext[k * 64 + t];
      Wf[k * 64 + t] = a[k] * w;
      acc = fmaf(c[k], w, acc);
    }
    bf[t] = acc;
  }
}

// ---------------- WMMA GEMM: Hout = relu(Hin @ Wf + bf), + BN stats --------
// Block = 256 threads = 8 waves; each wave owns 16 rows x 64 cols (4 C-tiles).
// In-place safe: a wave reads only its own 16 rows, then overwrites them.
__global__ __launch_bounds__(256) void ASAP_gemm_kernel(
    const float* __restrict__ Hin, float* __restrict__ Hout,
    const float* __restrict__ Wf, const float* __restrict__ bf,
    float* __restrict__ gSum, float* __restrict__ gSq) {
  __shared__ float Bt[64 * 64];   // W transposed: Bt[n][k]
  __shared__ float bias[64];
  __shared__ float sSum[64];
  __shared__ float sSq[64];
  int t = threadIdx.x;
  for (int i = t; i < 4096; i += 256) {
    int k = i >> 6, n = i & 63;
    Bt[n * 64 + k] = Wf[i];
  }
  if (t < 64) { bias[t] = bf[t]; sSum[t] = 0.0f; sSq[t] = 0.0f; }
  __syncthreads();

  const int wave = t >> 5;
  const int lane = t & 31;
  const int lhalf = lane >> 4;     // 0: lanes 0-15, 1: lanes 16-31
  const int l15 = lane & 15;
  const long rowBase = (long)blockIdx.x * 128 + wave * 16;

  // A 16x4 f32 frag: lane l<16 holds row l, k={k0,k0+1}; lane l+16 holds k={k0+2,k0+3}
  const float* Arow = Hin + (rowBase + l15) * 64 + 2 * lhalf;

  v8f c0 = {}, c1 = {}, c2 = {}, c3 = {};
  #pragma unroll
  for (int kk = 0; kk < 16; ++kk) {
    const int k0 = kk * 4;
    v2f a = *(const v2f*)(Arow + k0);
    const int kB = k0 + 2 * lhalf;   // B 4x16 frag: vgpr i, lane-half h -> k = k0+i+2h
    v2f b0 = *(const v2f*)(&Bt[(0 * 16 + l15) * 64 + kB]);
    v2f b1v = *(const v2f*)(&Bt[(1 * 16 + l15) * 64 + kB]);
    v2f b2v = *(const v2f*)(&Bt[(2 * 16 + l15) * 64 + kB]);
    v2f b3v = *(const v2f*)(&Bt[(3 * 16 + l15) * 64 + kB]);
    c0 = __builtin_amdgcn_wmma_f32_16x16x4_f32(false, a, false, b0,  (short)0, c0, false, false);
    c1 = __builtin_amdgcn_wmma_f32_16x16x4_f32(false, a, false, b1v, (short)0, c1, false, false);
    c2 = __builtin_amdgcn_wmma_f32_16x16x4_f32(false, a, false, b2v, (short)0, c2, false, false);
    c3 = __builtin_amdgcn_wmma_f32_16x16x4_f32(false, a, false, b3v, (short)0, c3, false, false);
  }

  // bias + ReLU + store + per-column stats.
  // C layout: vgpr r -> (M = r + 8*lhalf, N = l15 + 16*tile): all 8 values of a
  // lane share one column -> per-lane reduce, one LDS atomic per tile.
  v8f acc[4] = {c0, c1, c2, c3};
  #pragma unroll
  for (int tile = 0; tile < 4; ++tile) {
    const int n = tile * 16 + l15;
    const float bia = bias[n];
    float s = 0.0f, ss = 0.0f;
    #pragma unroll
    for (int r = 0; r < 8; ++r) {
      float v = fmaxf(acc[tile][r] + bia, 0.0f);
      s += v;
      ss = fmaf(v, v, ss);
      Hout[(rowBase + r + 8 * lhalf) * 64 + n] = v;
    }
    atomicAdd(&sSum[n], s);
    atomicAdd(&sSq[n], ss);
  }
  __syncthreads();
  if (t < 64) { atomicAdd(&gSum[t], sSum[t]); atomicAdd(&gSq[t], sSq[t]); }
}

// ---------------- scatter-max of BN3-normalized activations ----------------
__global__ __launch_bounds__(256) void ASAP_scatter_kernel(
    const float* __restrict__ Z, const int* __restrict__ ei,
    const float* __restrict__ a3, const float* __restrict__ c3,
    float* __restrict__ out, int E) {
  long tid = (long)blockIdx.x * 256 + threadIdx.x;
  long total = (long)E * 16;
  if (tid >= total) return;
  long e = tid >> 4;
  int q = (int)(tid & 15);
  int c0 = q * 4;
  int dst = ei[(long)E + e];
  float4 z = *(const float4*)(Z + e * 64 + c0);
  float* o = out + (long)dst * 64 + c0;
  atomic_max_f32(o + 0, fmaf(a3[c0 + 0], z.x, c3[c0 + 0]));
  atomic_max_f32(o + 1, fmaf(a3[c0 + 1], z.y, c3[c0 + 1]));
  atomic_max_f32(o + 2, fmaf(a3[c0 + 2], z.z, c3[c0 + 2]));
  atomic_max_f32(o + 3, fmaf(a3[c0 + 3], z.w, c3[c0 + 3]));
}

// ---------------- finalize: empty segments (-inf) -> 0 ---------------------
__global__ void ASAP_finalize_kernel(float* out, long n) {
  long tid = (long)blockIdx.x * blockDim.x + threadIdx.x;
  if (tid < n) {
    float v = out[tid];
    out[tid] = isfinite(v) ? v : 0.0f;
  }
}

// ---------------------------------------------------------------------------
extern "C" void kernel_launch(void* const* d_in, const int* in_sizes, int n_in,
                              void* d_out, int out_size, void* d_ws, size_t ws_size,
                              hipStream_t stream) {
  const float* x   = (const float*)d_in[0];
  const float* pos = (const float*)d_in[1];
  const int*   ei  = (const int*)d_in[2];
  const float* W1 = (const float*)d_in[3];
  const float* b1 = (const float*)d_in[4];
  const float* g1 = (const float*)d_in[5];
  const float* be1 = (const float*)d_in[6];
  const float* W2 = (const float*)d_in[7];
  const float* b2 = (const float*)d_in[8];
  const float* g2 = (const float*)d_in[9];
  const float* be2 = (const float*)d_in[10];
  const float* W3 = (const float*)d_in[11];
  const float* b3 = (const float*)d_in[12];
  const float* g3 = (const float*)d_in[13];
  const float* be3 = (const float*)d_in[14];
  float* out = (float*)d_out;

  const int E = in_sizes[2] / 2;           // edge_index is [2, E]
  const float invE = 1.0f / (float)E;

  // workspace layout (floats)
  float* w = (float*)d_ws;
  float* normsq = w + 0;
  float* sum1 = w + 64;   float* sq1 = w + 128;
  float* sum2 = w + 192;  float* sq2 = w + 256;
  float* sum3 = w + 320;  float* sq3 = w + 384;
  float* a3c = w + 448;   float* c3c = w + 512;
  float* W2f = w + 1024;  float* b2f = w + 5120;
  float* W3f = w + 6144;  float* b3f = w + 10240;
  float* H   = w + 16384;                  // E x 64 activations (in-place per layer)

  const long outN = (long)out_size;
  const int eBlocks = (E + 255) / 256;     // 6250
  const int gBlocks = (E + 127) / 128;     // 12500 (each block owns 128 rows)

  { // init
    long n = outN > 1024 ? outN : 1024;
    ASAP_init_kernel<<<(unsigned)((n + 255) / 256), 256, 0, stream>>>(w, out, outN);
  }
  ASAP_norm_kernel<<<eBlocks, 256, 0, stream>>>(pos, ei, normsq, E);
  ASAP_layer1_kernel<<<eBlocks, 256, 0, stream>>>(x, pos, ei, W1, b1, normsq,
                                                  H, sum1, sq1, E);
  ASAP_fold_kernel<<<1, 64, 0, stream>>>(sum1, sq1, g1, be1, W2, b2,
                                         W2f, b2f, nullptr, nullptr, invE);
  ASAP_gemm_kernel<<<gBlocks, 256, 0, stream>>>(H, H, W2f, b2f, sum2, sq2);
  ASAP_fold_kernel<<<1, 64, 0, stream>>>(sum2, sq2, g2, be2, W3, b3,
                                         W3f, b3f, nullptr, nullptr, invE);
  ASAP_gemm_kernel<<<gBlocks, 256, 0, stream>>>(H, H, W3f, b3f, sum3, sq3);
  ASAP_fold_kernel<<<1, 64, 0, stream>>>(sum3, sq3, g3, be3, nullptr, nullptr,
                                         nullptr, nullptr, a3c, c3c, invE);
  { // scatter-max: E*16 threads, each handles one (edge, 4-channel group)
    long total = (long)E * 16;
    ASAP_scatter_kernel<<<(unsigned)((total + 255) / 256), 256, 0, stream>>>(
        H, ei, a3c, c3c, out, E);
  }
  ASAP_finalize_kernel<<<(unsigned)((outN + 255) / 256), 256, 0, stream>>>(out, outN);
}